// SparseGatingNetwork_27900107554873
// MI455X (gfx1250) — compile-verified
//
#include <hip/hip_runtime.h>
#include <hip/hip_bf16.h>
#include <math.h>

// ---------------------------------------------------------------------------
// Sparse noisy top-k gating for MI455X (gfx1250).
// GEMM phase uses V_WMMA_F32_16X16X4_F32 (full fp32 precision, matrix pipe).
// ---------------------------------------------------------------------------

typedef float wmma_v2f __attribute__((ext_vector_type(2)));
typedef float wmma_v8f __attribute__((ext_vector_type(8)));

#define D_DIM      2048
#define E_DIM      64
#define NOISE_STD  0.1f
#define M_TILES    2                       // 16-row M tiles per wave
#define WAVES_PER_BLOCK 8
#define ROWS_PER_WAVE   (M_TILES * 16)     // 32
#define ROWS_PER_BLOCK  (WAVES_PER_BLOCK * ROWS_PER_WAVE) // 256

// ---------------------------------------------------------------------------
// Kernel 1: fused dual GEMM (gate + noise projections) + softplus/noise mix.
// Each wave computes a [32 x 64] tile of raw_gates.
// ---------------------------------------------------------------------------
__global__ __launch_bounds__(256) void gating_gemm_kernel(
    const float* __restrict__ x,        // [M, D]
    const float* __restrict__ gate_w,   // [E, D]
    const float* __restrict__ gate_b,   // [E]
    const float* __restrict__ noise_w,  // [E, D]
    const float* __restrict__ noise_b,  // [E]
    const float* __restrict__ noise,    // [M, E]
    float* __restrict__ raw_out)        // [M, E]
{
    const int lane   = threadIdx.x & 31;
    const int wave   = threadIdx.x >> 5;
    const int m_base = blockIdx.x * ROWS_PER_BLOCK + wave * ROWS_PER_WAVE;

    // WMMA f32 16x16x4 fragment addressing:
    //   A[m][k]: lane = (k>>1)*16 + m, vgpr = k&1  -> float2 @ x[row][k0 + 2*(lane>=16)]
    //   B[k][n]: lane = (k>>1)*16 + n, vgpr = k&1  -> float2 @ w[e ][k0 + 2*(lane>=16)]
    const int lm   = lane & 15;          // m (A) / n==e (B) within tile
    const int koff = (lane >> 4) << 1;   // 0 or 2

    wmma_v8f accG[M_TILES][4];
    wmma_v8f accN[M_TILES][4];
#pragma unroll
    for (int t = 0; t < M_TILES; ++t)
#pragma unroll
        for (int j = 0; j < 4; ++j) { accG[t][j] = {}; accN[t][j] = {}; }

    const float* xA[M_TILES];
#pragma unroll
    for (int t = 0; t < M_TILES; ++t)
        xA[t] = x + (size_t)(m_base + t * 16 + lm) * D_DIM + koff;

    const float* bg[4];
    const float* bn[4];
#pragma unroll
    for (int j = 0; j < 4; ++j) {
        bg[j] = gate_w  + (size_t)(j * 16 + lm) * D_DIM + koff;
        bn[j] = noise_w + (size_t)(j * 16 + lm) * D_DIM + koff;
    }

#pragma unroll 2
    for (int k0 = 0; k0 < D_DIM; k0 += 4) {
        wmma_v2f a[M_TILES];
#pragma unroll
        for (int t = 0; t < M_TILES; ++t)
            a[t] = *(const wmma_v2f*)(xA[t] + k0);

        wmma_v2f vg[4], vn[4];
#pragma unroll
        for (int j = 0; j < 4; ++j) {
            vg[j] = *(const wmma_v2f*)(bg[j] + k0);
            vn[j] = *(const wmma_v2f*)(bn[j] + k0);
        }

#pragma unroll
        for (int t = 0; t < M_TILES; ++t)
#pragma unroll
            for (int j = 0; j < 4; ++j) {
                accG[t][j] = __builtin_amdgcn_wmma_f32_16x16x4_f32(
                    false, a[t], false, vg[j], (short)0, accG[t][j], false, false);
                accN[t][j] = __builtin_amdgcn_wmma_f32_16x16x4_f32(
                    false, a[t], false, vn[j], (short)0, accN[t][j], false, false);
            }
    }

    // Epilogue: C/D layout -> vgpr r holds (M = r + 8*(lane>=16), N = lane&15).
    const int mhalf = (lane >> 4) << 3; // 0 or 8
#pragma unroll
    for (int t = 0; t < M_TILES; ++t) {
#pragma unroll
        for (int j = 0; j < 4; ++j) {
            const int   e  = j * 16 + lm;
            const float gb = gate_b[e];
            const float nb = noise_b[e];
            const float* ag = (const float*)&accG[t][j];
            const float* an = (const float*)&accN[t][j];
#pragma unroll
            for (int r = 0; r < 8; ++r) {
                const int row = m_base + t * 16 + r + mhalf;
                const float g  = ag[r] + gb;
                const float nl = an[r] + nb;
                // numerically stable softplus(nl) = max(nl,0) + log1p(exp(-|nl|))
                const float sp = fmaxf(nl, 0.0f) + log1pf(__expf(-fabsf(nl)));
                const float nz = noise[(size_t)row * E_DIM + e];
                raw_out[(size_t)row * E_DIM + e] = fmaf(nz * NOISE_STD, sp, g);
            }
        }
    }
}

// ---------------------------------------------------------------------------
// Kernel 2: per-row top-2 (first-index-wins on ties, matching jax.lax.top_k)
// followed by a 2-way softmax.
// ---------------------------------------------------------------------------
__global__ __launch_bounds__(256) void topk_softmax_kernel(
    const float* __restrict__ raw,   // [M, E]
    float* __restrict__ gates,       // [M, 2]
    int*   __restrict__ idx,         // [M, 2]
    int n_rows)
{
    const int row = blockIdx.x * blockDim.x + threadIdx.x;
    if (row >= n_rows) return;

    const float* r = raw + (size_t)row * E_DIM;
    float v1 = -INFINITY, v2 = -INFINITY;
    int   i1 = 0,         i2 = 0;
#pragma unroll 8
    for (int e = 0; e < E_DIM; ++e) {
        const float v = r[e];
        if (v > v1)      { v2 = v1; i2 = i1; v1 = v; i1 = e; }
        else if (v > v2) { v2 = v;  i2 = e; }
    }
    // softmax over {v1, v2} with v1 >= v2
    const float ex = __expf(v2 - v1);
    const float g1 = 1.0f / (1.0f + ex);
    const float g2 = ex * g1;

    gates[(size_t)row * 2 + 0] = g1;
    gates[(size_t)row * 2 + 1] = g2;
    idx[(size_t)row * 2 + 0]   = i1;
    idx[(size_t)row * 2 + 1]   = i2;
}

// ---------------------------------------------------------------------------
extern "C" void kernel_launch(void* const* d_in, const int* in_sizes, int n_in,
                              void* d_out, int out_size, void* d_ws, size_t ws_size,
                              hipStream_t stream)
{
    const float* x       = (const float*)d_in[0];
    const float* gate_w  = (const float*)d_in[1];
    const float* gate_b  = (const float*)d_in[2];
    const float* noise_w = (const float*)d_in[3];
    const float* noise_b = (const float*)d_in[4];
    const float* noise   = (const float*)d_in[5];

    const int M = in_sizes[0] / D_DIM;   // B*S = 32768

    // d_out layout (flat, return order): gates [M,2] f32, indices [M,2] i32,
    // raw_gates [M,64] f32.
    float* out    = (float*)d_out;
    float* gates  = out;
    int*   idx    = (int*)(out + (size_t)M * 2);
    float* raw    = out + (size_t)M * 4;

    dim3 blk(256);
    dim3 grid1(M / ROWS_PER_BLOCK);
    gating_gemm_kernel<<<grid1, blk, 0, stream>>>(x, gate_w, gate_b,
                                                  noise_w, noise_b, noise, raw);

    dim3 grid2((M + 255) / 256);
    topk_softmax_kernel<<<grid2, blk, 0, stream>>>(raw, gates, idx, M);
}